// SmoothedCrossEntropyLoss_79894981640438
// MI455X (gfx1250) — compile-verified
//
#include <hip/hip_runtime.h>
#include <stdint.h>

// Problem constants (match reference)
constexpr int   kB    = 2;
constexpr int   kC    = 32000;
constexpr int   kS    = 2048;
constexpr int   kRows = kB * kS;          // 4096
constexpr int   kIgnore = -100;

// Tiling
constexpr int   TILE_S = 256;             // s-positions per block (contiguous in memory)
constexpr int   TILE_C = 10;              // classes per TDM tile -> 10 KB tile, 20 KB double-buffered

constexpr float kLog2e  = 1.4426950408889634f;
constexpr float kLn2    = 0.6931471805599453f;
constexpr float kNegMax = -3.402823466e38f;

typedef uint32_t u32x4 __attribute__((ext_vector_type(4)));
typedef uint32_t u32x8 __attribute__((ext_vector_type(8)));
typedef float    v8f   __attribute__((ext_vector_type(8)));
typedef float    v2f   __attribute__((ext_vector_type(2)));

// Low 32 bits of a flat shared-memory address are the LDS byte offset
// (LDS aperture; CDNA5 ISA).
__device__ __forceinline__ uint32_t lds_offset(const void* p) {
  return (uint32_t)(uintptr_t)p;
}

// Issue one TDM 2D tile load: tile_dim0 = TILE_S contiguous fp32 along s,
// tile_dim1 = TILE_C classes, row stride = kS elements. Tensor dims == tile
// dims so the transfer is always in-bounds. D# layout per CDNA5 ISA.
__device__ __forceinline__ void tdm_load_tile(const float* gaddr, uint32_t lds) {
  uint64_t ga = (uint64_t)(uintptr_t)gaddr;
  u32x4 g0;
  g0[0] = 1u;                                              // count=1 (valid), user mode
  g0[1] = lds;                                             // lds_addr [63:32]
  g0[2] = (uint32_t)ga;                                    // global_addr lo
  g0[3] = (uint32_t)((ga >> 32) & 0x01FFFFFFu) | (2u << 30); // addr hi | type=2
  u32x8 g1;
  g1[0] = 2u << 16;                                        // data_size=2 (4 bytes)
  g1[1] = (uint32_t)TILE_S << 16;                          // tensor_dim0[15:0] @ bits[63:48]
  g1[2] = (uint32_t)TILE_C << 16;                          // tensor_dim1[15:0] @ bits[95:80]
  g1[3] = (uint32_t)TILE_S << 16;                          // tile_dim0 @ bits[127:112]
  g1[4] = (uint32_t)TILE_C;                                // tile_dim1 @ bits[143:128]
  g1[5] = (uint32_t)kS;                                    // tensor_dim0_stride lo32
  g1[6] = 0u;
  g1[7] = 0u;
  asm volatile("tensor_load_to_lds %0, %1" :: "s"(g0), "s"(g1) : "memory");
}

// ---------------- Stage 1: per-(row, class-chunk) online-softmax partials ----------------
__global__ __launch_bounds__(TILE_S) void sce_stage1(const float* __restrict__ x,
                                                     float4* __restrict__ part,
                                                     int chunkC) {
  __shared__ float tile[2][TILE_C * TILE_S];
  const int tid   = threadIdx.x;
  const int sb    = blockIdx.x;                  // 16 blocks over (b, s0)
  const int chunk = blockIdx.y;
  const int b     = sb / (kS / TILE_S);
  const int s0    = (sb % (kS / TILE_S)) * TILE_S;
  const int c0    = chunk * chunkC;
  const int ntiles = chunkC / TILE_C;
  const bool issuer = (tid < 32);                // wave 0 owns TDM issue + TENSORcnt

  const float* base = x + (size_t)b * kC * kS + (size_t)c0 * kS + s0;

  if (issuer) {
    tdm_load_tile(base, lds_offset(&tile[0][0]));
    if (ntiles > 1) tdm_load_tile(base + (size_t)TILE_C * kS, lds_offset(&tile[1][0]));
  }

  // Accumulators in base-2 domain: m = max(x*log2e), sum = sum 2^(y-m)
  float m = kNegMax, sum = 0.f, fsum = 0.f, cnt = 0.f;

  for (int t = 0; t < ntiles; ++t) {
    if (issuer) {
      if (t < ntiles - 1) __builtin_amdgcn_s_wait_tensorcnt(1);  // tile t done (in-order)
      else                __builtin_amdgcn_s_wait_tensorcnt(0);
    }
    __syncthreads();                             // tile t visible to all 8 waves
    const float* buf = &tile[t & 1][0];
#pragma unroll
    for (int c = 0; c < TILE_C; ++c) {
      float v = buf[c * TILE_S + tid];           // ds_load_b32, conflict-free
      float y = v * kLog2e;
      float d = y - m;
      bool  gt = d > 0.f;
      m = fmaxf(m, y);
      float p  = __builtin_amdgcn_exp2f(-__builtin_fabsf(d));  // exp2(min-max); -inf -> 0
      float sn = sum + p;                        // v <= m path
      float sy = __builtin_fmaf(sum, p, 1.f);    // new-max path: sum*2^(m-y)+1
      sum = gt ? sy : sn;
      bool fin = v >= kNegMax;                   // finite (inputs are finite or -inf)
      fsum += fin ? v : 0.f;
      cnt  += fin ? 1.f : 0.f;
    }
    __syncthreads();                             // all waves done reading buf
    if (issuer && (t + 2) < ntiles)
      tdm_load_tile(base + (size_t)(t + 2) * TILE_C * kS, lds_offset(&tile[t & 1][0]));
  }

  const int row = sb * TILE_S + tid;             // row = b*S + s
  part[(size_t)chunk * kRows + row] = make_float4(m, sum, fsum, cnt);
}

// ---------------- Stage 2: merge chunk partials per row, add target term ----------------
__global__ __launch_bounds__(256) void sce_stage2(const float* __restrict__ x,
                                                  const int* __restrict__ tgt,
                                                  const float4* __restrict__ part,
                                                  float* __restrict__ lossbuf,
                                                  int chunks) {
  const int row = blockIdx.x * 256 + threadIdx.x;
  float M = kNegMax, Ssum = 0.f, fsum = 0.f, cnt = 0.f;
  for (int k = 0; k < chunks; ++k) {
    float4 p = part[(size_t)k * kRows + row];
    float M2 = fmaxf(M, p.x);
    float sA = __builtin_amdgcn_exp2f(M - M2);
    float sB = __builtin_amdgcn_exp2f(p.x - M2);
    Ssum = __builtin_fmaf(Ssum, sA, p.y * sB);
    M = M2;
    fsum += p.z;
    cnt  += p.w;
  }
  const float lse = kLn2 * (M + __builtin_amdgcn_logf(Ssum));  // natural-log LSE

  const int t = tgt[row];
  float loss = 0.f;
  if (t != kIgnore) {
    const int b = row / kS;
    const int s = row % kS;
    const float xt = x[((size_t)b * kC + (size_t)t) * kS + s];
    float l = 0.1f * lse - (0.1f / cnt) * fsum;                // smoothing term
    if (xt >= kNegMax) l += 0.9f * (lse - xt);                 // target term (finite only)
    loss = l;
  }
  lossbuf[row] = loss;
}

// ---------------- Stage 3: reduce 4096 row losses -> scalar (WMMA ones-matrix sum) ------
__global__ __launch_bounds__(256) void sce_stage3(const float* __restrict__ lossbuf,
                                                  float* __restrict__ out) {
  __shared__ float red[256];
  const int tid = threadIdx.x;
  float a = 0.f;
  for (int i = tid; i < kRows; i += 256) a += lossbuf[i];
  red[tid] = a;
  __syncthreads();
#if __has_builtin(__builtin_amdgcn_wmma_f32_16x16x4_f32)
  if (tid < 32) {                                  // wave 0, EXEC all ones
    v2f onesv; onesv[0] = 1.f; onesv[1] = 1.f;     // B = ones(4x16)
    v8f c = {};
#pragma unroll
    for (int t = 0; t < 4; ++t) {                  // A holds 64 values per wmma
      v2f av;
      av[0] = red[t * 64 + tid * 2];
      av[1] = red[t * 64 + tid * 2 + 1];
      c = __builtin_amdgcn_wmma_f32_16x16x4_f32(false, av, false, onesv,
                                                (short)0, c, false, false);
    }
    float r = 0.f;
#pragma unroll
    for (int k = 0; k < 8; ++k) r += c[k];         // sum half of D's rows per lane group
    r += __shfl_xor(r, 16, 32);                    // combine the two half-lane groups
    if (tid == 0) out[0] = r * (1.0f / kRows);     // batchmean: / (B*S)
  }
#else
  for (int sft = 128; sft > 0; sft >>= 1) {
    if (tid < sft) red[tid] += red[tid + sft];
    __syncthreads();
  }
  if (tid == 0) out[0] = red[0] * (1.0f / kRows);
#endif
}

// ---------------- Launch ----------------
extern "C" void kernel_launch(void* const* d_in, const int* in_sizes, int n_in,
                              void* d_out, int out_size, void* d_ws, size_t ws_size,
                              hipStream_t stream) {
  (void)in_sizes; (void)n_in; (void)out_size;
  const float* x   = (const float*)d_in[0];
  const int*   tgt = (const int*)d_in[1];
  float*       out = (float*)d_out;

  // Pick class-chunk count that fits the workspace (64 -> 4 MB partials).
  int chunks = 64;
  while (chunks > 1 &&
         ((size_t)chunks * kRows * sizeof(float4) + (size_t)kRows * sizeof(float)) > ws_size)
    chunks >>= 1;
  const int chunkC = kC / chunks;                  // 500..32000, all multiples of TILE_C

  float4* part    = (float4*)d_ws;
  float*  lossbuf = (float*)((char*)d_ws + (size_t)chunks * kRows * sizeof(float4));

  dim3 grid1(kRows / TILE_S, chunks);              // 16 x 64 = 1024 blocks, 8 waves each
  sce_stage1<<<grid1, TILE_S, 0, stream>>>(x, part, chunkC);
  sce_stage2<<<kRows / 256, 256, 0, stream>>>(x, tgt, part, lossbuf, chunks);
  sce_stage3<<<1, 256, 0, stream>>>(lossbuf, out);
}